// PhysicsInformedLayer_41635412967599
// MI455X (gfx1250) — compile-verified
//
#include <hip/hip_runtime.h>
#include <hip/hip_bf16.h>
#include <math.h>

// ---------------------------------------------------------------------------
// PhysicsInformedLayer for MI455X (gfx1250, wave32)
//   Phase 1: radii = 0.01 * softplus(MLP(features))   -- fp32 WMMA 16x16x4
//   Phase 2: N^2 pairwise repulsion forces            -- LDS-tiled VALU loop
// ---------------------------------------------------------------------------

typedef __attribute__((ext_vector_type(2))) float v2f;
typedef __attribute__((ext_vector_type(8))) float v8f;

#define N_PTS  4096
#define FEAT_K 256
#define H1_N   128
#define H2_N   64
#define EPS_F  1e-6f

// ===========================================================================
// Kernel 1: fused MLP -> radii.  One block = 16 samples, 256 threads = 8 waves.
//
// WMMA fp32 fragment layouts (CDNA5 ISA 7.12.2, 32-bit data):
//   A (16x4, MxK):  lanes 0-15 hold M=0..15 with VGPR0=K0,VGPR1=K1;
//                   lanes 16-31 hold M=0..15 with VGPR0=K2,VGPR1=K3.
//   B (4x16, KxN):  row striped across lanes within a VGPR; rows split in
//                   lane-halves mirroring A:  VGPR0 = {K0 | K2}, VGPR1 = {K1 | K3}.
//   C/D (16x16):    VGPR v: lanes 0-15 -> M=v, lanes 16-31 -> M=v+8, N=lane%16.
// ===========================================================================
__global__ __launch_bounds__(256)
void mlp_radii_kernel(const float* __restrict__ feat,
                      const float* __restrict__ W1, const float* __restrict__ b1,
                      const float* __restrict__ W2, const float* __restrict__ b2,
                      const float* __restrict__ W3, const float* __restrict__ b3,
                      float* __restrict__ radii_out)
{
    __shared__ float h1s[16 * H1_N];   // 8 KB
    __shared__ float h2s[16 * H2_N];   // 4 KB

    const int tid  = threadIdx.x;
    const int wave = tid >> 5;         // 0..7 (wave32)
    const int lane = tid & 31;
    const int half = lane >> 4;        // 0: lanes 0-15, 1: lanes 16-31
    const int r    = lane & 15;
    const int row0 = blockIdx.x * 16;  // 16 samples per block

    // ---- layer 1: h1[16x128] = relu(feat[16x256] @ W1[256x128] + b1) ------
    {
        const int col = wave * 16 + r;                 // each wave: one 16-col tile
        const float* arow = feat + (long)(row0 + r) * FEAT_K;
        v8f c = {};
        for (int k0 = 0; k0 < FEAT_K; k0 += 4) {
            v2f a, b;
            a.x = arow[k0 + 2 * half + 0];
            a.y = arow[k0 + 2 * half + 1];
            b.x = W1[(k0 + 2 * half + 0) * H1_N + col];
            b.y = W1[(k0 + 2 * half + 1) * H1_N + col];
            c = __builtin_amdgcn_wmma_f32_16x16x4_f32(
                    false, a, false, b, (short)0, c, false, false);
        }
        const float bias = b1[col];
#pragma unroll
        for (int v = 0; v < 8; ++v) {
            const int m = v + 8 * half;
            const float val = c[v] + bias;
            h1s[m * H1_N + col] = val > 0.f ? val : 0.f;
        }
    }
    __syncthreads();

    // ---- layer 2: h2[16x64] = relu(h1 @ W2[128x64] + b2), waves 0-3 -------
    if (wave < 4) {                                    // wave-uniform branch: EXEC stays all-1s
        const int col = wave * 16 + r;
        v8f c = {};
        for (int k0 = 0; k0 < H1_N; k0 += 4) {
            v2f a, b;
            a.x = h1s[r * H1_N + k0 + 2 * half + 0];
            a.y = h1s[r * H1_N + k0 + 2 * half + 1];
            b.x = W2[(k0 + 2 * half + 0) * H2_N + col];
            b.y = W2[(k0 + 2 * half + 1) * H2_N + col];
            c = __builtin_amdgcn_wmma_f32_16x16x4_f32(
                    false, a, false, b, (short)0, c, false, false);
        }
        const float bias = b2[col];
#pragma unroll
        for (int v = 0; v < 8; ++v) {
            const int m = v + 8 * half;
            const float val = c[v] + bias;
            h2s[m * H2_N + col] = val > 0.f ? val : 0.f;
        }
    }
    __syncthreads();

    // ---- layer 3: radii = 0.01 * softplus(h2 @ W3 + b3) -------------------
    if (tid < 16) {
        float acc = b3[0];
#pragma unroll 8
        for (int k = 0; k < H2_N; ++k)
            acc += h2s[tid * H2_N + k] * W3[k];
        const float sp = (acc > 20.f) ? acc : log1pf(expf(acc));
        radii_out[row0 + tid] = 0.01f * sp;
    }
}

// ===========================================================================
// Kernel 2: pairwise repulsion forces.  One thread per i, j tiled via LDS.
//   overlap = relu(2*radii[j] - dist(i,j) + eps) * (i!=j)
//   f_i    += overlap * (1 + 0.5*[type_i==type_j]) * (x_i - x_j)/(dist+eps)
// ===========================================================================
#define JTILE 256
__global__ __launch_bounds__(256)
void forces_kernel(const float* __restrict__ coords,
                   const int*   __restrict__ ctype,
                   const float* __restrict__ radii,
                   const float* __restrict__ rstrength,
                   float* __restrict__ forces)
{
    __shared__ float sx[JTILE], sy[JTILE], sr2[JTILE];
    __shared__ int   st[JTILE];

    const int i  = blockIdx.x * blockDim.x + threadIdx.x;
    const float xi = coords[2 * i + 0];
    const float yi = coords[2 * i + 1];
    const int   ti = ctype[i];

    float fx = 0.f, fy = 0.f;

    for (int j0 = 0; j0 < N_PTS; j0 += JTILE) {
        __syncthreads();
        const int j = j0 + threadIdx.x;
        sx[threadIdx.x]  = coords[2 * j + 0];
        sy[threadIdx.x]  = coords[2 * j + 1];
        sr2[threadIdx.x] = 2.f * radii[j];   // radii[j] + radii[j] (column broadcast)
        st[threadIdx.x]  = ctype[j];
        __syncthreads();

        for (int jj = 0; jj < JTILE; ++jj) {
            const float dx = xi - sx[jj];
            const float dy = yi - sy[jj];
            const float sq = dx * dx + dy * dy;
            const float dist = sqrtf(sq);           // exact 0 on diagonal; masked below
            float ov = sr2[jj] - dist + EPS_F;
            ov = ov > 0.f ? ov : 0.f;
            const float w = (ti == st[jj]) ? 1.5f : 1.0f;
            const float s = ((j0 + jj) == i) ? 0.f : ov * w / (dist + EPS_F);
            fx += s * dx;
            fy += s * dy;
        }
    }

    const float k = rstrength[0];
    forces[2 * i + 0] = fx * k;
    forces[2 * i + 1] = fy * k;
}

// ===========================================================================
extern "C" void kernel_launch(void* const* d_in, const int* in_sizes, int n_in,
                              void* d_out, int out_size, void* d_ws, size_t ws_size,
                              hipStream_t stream)
{
    const float* coords = (const float*)d_in[0];   // [N,2]
    const float* feats  = (const float*)d_in[1];   // [N,256]
    const int*   ctype  = (const int*)  d_in[2];   // [N]
    const float* W1     = (const float*)d_in[3];   // [256,128]
    const float* b1     = (const float*)d_in[4];   // [128]
    const float* W2     = (const float*)d_in[5];   // [128,64]
    const float* b2     = (const float*)d_in[6];   // [64]
    const float* W3     = (const float*)d_in[7];   // [64,1]
    const float* b3     = (const float*)d_in[8];   // [1]
    const float* rs     = (const float*)d_in[9];   // scalar

    float* out    = (float*)d_out;
    float* forces = out;               // [N,2] first in tuple order
    float* radii  = out + 2 * N_PTS;   // [N]   second

    mlp_radii_kernel<<<N_PTS / 16, 256, 0, stream>>>(feats, W1, b1, W2, b2, W3, b3, radii);
    forces_kernel<<<N_PTS / 256, 256, 0, stream>>>(coords, ctype, radii, rs, forces);

    (void)in_sizes; (void)n_in; (void)out_size; (void)d_ws; (void)ws_size;
}